// FABindProteinComplex_27109833572511
// MI455X (gfx1250) — compile-verified
//
#include <hip/hip_runtime.h>
#include <math.h>

typedef __attribute__((ext_vector_type(2))) float v2f;
typedef __attribute__((ext_vector_type(8))) float v8f;

// Problem constants
#define BB    16
#define NCC   128
#define NPP   256
#define CC    128
#define HIDD  256
#define MAXNN 386   // 2 + NC + NP

// Output float offsets
#define O_CCOORD   0        // 6144
#define O_CBATCH   6144     // 2048
#define O_PCOORD   8192     // 12288
#define O_PBATCH   20480    // 4096
#define O_YPRED    24576    // 524288
#define O_YCOORD   548864   // 524288
#define O_DISMAP   1073152  // 524288
#define O_KEEPLT5  1597440  // 1

// ---------------------------------------------------------------------------
// WMMA f32 GEMM: C[M,N] = (gather?)A[M,K] @ B[K,N] (+ bias[N])
// A,B,C row-major. One wave computes a 16x64 tile (4 independent WMMA
// accumulator chains). Requirements: K % 4 == 0, N % 64 == 0.
// A row index clamped (not guarded) -> no divergent loads; stores guarded.
// ---------------------------------------------------------------------------
__global__ void gemm_wmma_f32(const float* __restrict__ A, int lda,
                              const int* __restrict__ ridx,
                              const float* __restrict__ Bm, int ldb,
                              float* __restrict__ Cm, int ldc,
                              const float* __restrict__ bias,
                              int M, int N, int K)
{
  const int wave = threadIdx.x >> 5;
  const int lane = threadIdx.x & 31;
  const int tilesN = N >> 6;                 // 64-wide N tiles
  const int tilesM = (M + 15) >> 4;
  const int tile = blockIdx.x * (blockDim.x >> 5) + wave;
  if (tile >= tilesM * tilesN) return;       // wave-uniform exit
  const int tm = tile / tilesN;
  const int tn = tile % tilesN;

  const int lrow  = lane & 15;
  const int khalf = (lane >> 4) << 1;        // 0 (lanes 0-15) or 2 (16-31)
  int arow = tm * 16 + lrow;
  if (arow >= M) arow = M - 1;               // clamp: no divergent loads
  const long asrc = ridx ? (long)ridx[arow] : (long)arow;
  const int col = tn * 64 + lrow;

  const float* __restrict__ Ap = A + asrc * (long)lda + khalf;
  const float* __restrict__ Bp = Bm + (long)khalf * ldb + col;
  const long bstep = 4l * ldb;

  v8f acc0 = {}, acc1 = {}, acc2 = {}, acc3 = {};
  #pragma unroll 2
  for (int k = 0; k < K; k += 4) {
    // A 16x4 fragment: two consecutive floats per lane -> single b64 load
    const v2f a = *(const v2f*)Ap;
    // B 4x16 fragments for 4 N-subtiles
    const v2f b0 = { Bp[0],  Bp[ldb]      };
    const v2f b1 = { Bp[16], Bp[ldb + 16] };
    const v2f b2 = { Bp[32], Bp[ldb + 32] };
    const v2f b3 = { Bp[48], Bp[ldb + 48] };
    acc0 = __builtin_amdgcn_wmma_f32_16x16x4_f32(false, a, false, b0, (short)0, acc0, false, false);
    acc1 = __builtin_amdgcn_wmma_f32_16x16x4_f32(false, a, false, b1, (short)0, acc1, false, false);
    acc2 = __builtin_amdgcn_wmma_f32_16x16x4_f32(false, a, false, b2, (short)0, acc2, false, false);
    acc3 = __builtin_amdgcn_wmma_f32_16x16x4_f32(false, a, false, b3, (short)0, acc3, false, false);
    Ap += 4;
    Bp += bstep;
  }

  const int rbase = tm * 16 + ((lane >> 4) << 3);  // rows 0..7 or 8..15
  float bv0 = 0.f, bv1 = 0.f, bv2 = 0.f, bv3 = 0.f;
  if (bias) {
    bv0 = bias[col];
    bv1 = bias[col + 16];
    bv2 = bias[col + 32];
    bv3 = bias[col + 48];
  }
  #pragma unroll
  for (int r = 0; r < 8; ++r) {
    const int row = rbase + r;
    if (row < M) {
      float* __restrict__ Crow = Cm + (long)row * ldc + col;
      Crow[0]  = acc0[r] + bv0;
      Crow[16] = acc1[r] + bv1;
      Crow[32] = acc2[r] + bv2;
      Crow[48] = acc3[r] + bv3;
    }
  }
}

// ---------------------------------------------------------------------------
// bp1[k] = b1[k] + sum_c b_prot[c]*W1[c,k] ; bc1[k] = sum_c b_comp[c]*W1[c,k]
// ---------------------------------------------------------------------------
__global__ void bias_prep(const float* __restrict__ b_prot,
                          const float* __restrict__ b_comp,
                          const float* __restrict__ W1,
                          const float* __restrict__ b1,
                          float* __restrict__ bp1, float* __restrict__ bc1)
{
  const int k = threadIdx.x;  // 256 threads
  float sp = b1[k], sc = 0.0f;
  for (int c = 0; c < CC; ++c) {
    const float w = W1[c * HIDD + k];
    sp = fmaf(b_prot[c], w, sp);
    sc = fmaf(b_comp[c], w, sc);
  }
  bp1[k] = sp;
  bc1[k] = sc;
}

// ---------------------------------------------------------------------------
// Pair eval: y[b,i,j] = 10*sigmoid(b2 + sum_k relu(P1[b,i,k]+C1[b,j,k])*W2[k])
// Block = 256 threads handles (b, 16 i's, 32 j's). LDS stride 257 -> no bank
// conflicts (j = lane, stride 257 dwords).
// ---------------------------------------------------------------------------
__global__ void pair_eval(const float* __restrict__ P1,
                          const float* __restrict__ C1,
                          const float* __restrict__ W2,
                          const float* __restrict__ b2,
                          float* __restrict__ ypred)
{
  __shared__ float sP[16 * 257];
  __shared__ float sC[32 * 257];
  __shared__ float sW[HIDD];

  const int blk   = blockIdx.x;        // b*64 + iTile*4 + jTile
  const int jTile = blk & 3;
  const int iTile = (blk >> 2) & 15;
  const int b     = blk >> 6;
  const int tid   = threadIdx.x;

  for (int t = tid; t < 16 * HIDD; t += 256) {
    const int i = t >> 8, k = t & 255;
    sP[i * 257 + k] = P1[((long)(b * NPP + iTile * 16 + i)) * HIDD + k];
  }
  for (int t = tid; t < 32 * HIDD; t += 256) {
    const int j = t >> 8, k = t & 255;
    sC[j * 257 + k] = C1[((long)(b * NCC + jTile * 32 + j)) * HIDD + k];
  }
  sW[tid] = W2[tid];
  __syncthreads();

  const int lane = tid & 31;
  const int w    = tid >> 5;   // 0..7 -> i offsets w and w+8
  const float* __restrict__ p0 = &sP[w * 257];
  const float* __restrict__ p1 = &sP[(w + 8) * 257];
  const float* __restrict__ cj = &sC[lane * 257];

  float acc0 = 0.0f, acc1 = 0.0f;
  for (int k = 0; k < HIDD; ++k) {
    const float cv = cj[k];
    const float wv = sW[k];
    float h0 = p0[k] + cv; h0 = h0 > 0.0f ? h0 : 0.0f;
    float h1 = p1[k] + cv; h1 = h1 > 0.0f ? h1 : 0.0f;
    acc0 = fmaf(h0, wv, acc0);
    acc1 = fmaf(h1, wv, acc1);
  }
  const float bb = b2[0];
  const float y0 = 10.0f / (1.0f + expf(-(acc0 + bb)));
  const float y1 = 10.0f / (1.0f + expf(-(acc1 + bb)));
  const int i0 = iTile * 16 + w;
  const int i1 = i0 + 8;
  const int j  = jTile * 32 + lane;
  ypred[((long)(b * NPP + i0)) * NCC + j] = y0;
  ypred[((long)(b * NPP + i1)) * NCC + j] = y1;
}

// ---------------------------------------------------------------------------
// Distance map (clip(||p/5 - c/5||*5, 0, 10)) + dis_map passthrough
// ---------------------------------------------------------------------------
__global__ void dmap_copy(const float* __restrict__ coords,
                          const float* __restrict__ dismap,
                          float* __restrict__ out)
{
  const int idx = blockIdx.x * 256 + threadIdx.x;  // < 524288
  const int j = idx & 127;
  const int i = (idx >> 7) & 255;
  const int b = idx >> 15;
  const float* pc = coords + ((long)(b * MAXNN + 2 + NCC + i)) * 3;
  const float* cc = coords + ((long)(b * MAXNN + 1 + j)) * 3;
  float d2 = 1e-12f;
  #pragma unroll
  for (int d = 0; d < 3; ++d) {
    const float dd = pc[d] * 0.2f - cc[d] * 0.2f;
    d2 = fmaf(dd, dd, d2);
  }
  float v = sqrtf(d2) * 5.0f;
  v = fminf(fmaxf(v, 0.0f), 10.0f);
  out[O_YCOORD + idx] = v;
  out[O_DISMAP + idx] = dismap[idx];
}

// ---------------------------------------------------------------------------
// Coords outputs, batch index outputs, scalar output
// ---------------------------------------------------------------------------
__global__ void misc_out(const float* __restrict__ coords, float* __restrict__ out)
{
  const int tid = blockIdx.x * 256 + threadIdx.x;  // < 12288
  if (tid < BB * NCC * 3) {                        // compound coords (round-trip /5*5)
    const int r = tid / 3, d = tid % 3;
    const int b = r >> 7, j = r & 127;
    out[O_CCOORD + tid] = (coords[((long)(b * MAXNN + 1 + j)) * 3 + d] / 5.0f) * 5.0f;
  }
  if (tid < BB * NCC)  out[O_CBATCH + tid] = (float)(tid >> 7);
  if (tid < BB * NPP * 3) {                        // pocket coords
    const int r = tid / 3, d = tid % 3;
    const int b = r >> 8, i = r & 255;
    out[O_PCOORD + tid] = (coords[((long)(b * MAXNN + 2 + NCC + i)) * 3 + d] / 5.0f) * 5.0f;
  }
  if (tid < BB * NPP)  out[O_PBATCH + tid] = (float)(tid >> 8);
  if (tid == 0)        out[O_KEEPLT5] = 0.0f;
}

// ---------------------------------------------------------------------------
extern "C" void kernel_launch(void* const* d_in, const int* in_sizes, int n_in,
                              void* d_out, int out_size, void* d_ws, size_t ws_size,
                              hipStream_t stream)
{
  const float* protein  = (const float*)d_in[0];   // 16384 x 1280
  const float* compound = (const float*)d_in[1];   // 2048 x 56
  const float* coords   = (const float*)d_in[2];   // B*386 x 3
  const float* dismap   = (const float*)d_in[3];   // 524288
  const int*   keep     = (const int*)d_in[4];     // 4096
  // d_in[5], d_in[6]: glb_c / glb_p -- sliced away by pair[:,1:,1:,:], unused
  const float* W_prot   = (const float*)d_in[7];   // 1280 x 128
  const float* b_prot   = (const float*)d_in[8];   // 128
  const float* W_comp   = (const float*)d_in[9];   // 56 x 128
  const float* b_comp   = (const float*)d_in[10];  // 128
  const float* W1       = (const float*)d_in[11];  // 128 x 256
  const float* b1       = (const float*)d_in[12];  // 256
  const float* W2       = (const float*)d_in[13];  // 256 x 1
  const float* b2       = (const float*)d_in[14];  // 1

  float* out = (float*)d_out;
  float* ws  = (float*)d_ws;

  float* Wp1 = ws;                       // 1280*256
  float* Wc1 = Wp1 + 1280 * 256;         // 56*256
  float* bp1 = Wc1 + 56 * 256;           // 256
  float* bc1 = bp1 + 256;                // 256
  float* P1  = bc1 + 256;                // 4096*256
  float* C1  = P1 + 4096 * 256;          // 2048*256  (total ~7.3 MB)

  // Fused weights: Wp1 = W_prot @ W1 (M=1280,N=256,K=128): 80*4=320 tiles
  gemm_wmma_f32<<<40, 256, 0, stream>>>(W_prot, 128, nullptr, W1, 256,
                                        Wp1, 256, nullptr, 1280, 256, 128);
  // Wc1 = W_comp @ W1 (M=56,N=256,K=128): 4*4=16 tiles
  gemm_wmma_f32<<<2, 256, 0, stream>>>(W_comp, 128, nullptr, W1, 256,
                                       Wc1, 256, nullptr, 56, 256, 128);
  // Fused biases
  bias_prep<<<1, 256, 0, stream>>>(b_prot, b_comp, W1, b1, bp1, bc1);

  // P1 = protein[keep] @ Wp1 + bp1 (M=4096,N=256,K=1280): 256*4=1024 tiles
  gemm_wmma_f32<<<128, 256, 0, stream>>>(protein, 1280, keep, Wp1, 256,
                                         P1, 256, bp1, 4096, 256, 1280);
  // C1 = compound @ Wc1 + bc1 (M=2048,N=256,K=56): 128*4=512 tiles
  gemm_wmma_f32<<<64, 256, 0, stream>>>(compound, 56, nullptr, Wc1, 256,
                                        C1, 256, bc1, 2048, 256, 56);

  // Pair MLP evaluation (relu-dot, factorized)
  pair_eval<<<BB * 16 * 4, 256, 0, stream>>>(P1, C1, W2, b2, out + O_YPRED);

  // Distance map + dis_map passthrough (524288 elems)
  dmap_copy<<<2048, 256, 0, stream>>>(coords, dismap, out);

  // Coords / batch / scalar outputs (12288 threads)
  misc_out<<<48, 256, 0, stream>>>(coords, out);
}